// ExpertParallelBehindBlockDS_2834678415772
// MI455X (gfx1250) — compile-verified
//
#include <hip/hip_runtime.h>

// ---------------------------------------------------------------------------
// ExpertParallelBehindBlockDS for MI455X (gfx1250), fp32 WMMA path.
//
// 3 GEMMs, all fp32 (memory-bound: ~520MB traffic @ 23.3TB/s >> WMMA time,
// so V_WMMA_F32_16X16X4_F32 keeps full precision for free):
//   1) expert_out[e] = inputs[e*C : ] (1024x4096) @ expert_w[e] (4096x1024) + b
//   2) combined      = combine_w (4096x8192) @ expert_out (8192x1024) -> d_out
//   3) d_out         = combined*w0 + (resid_in @ residual_w + rb)*w1
//
// Tile staging uses CDNA5 async global->LDS DMA (GLOBAL_LOAD_ASYNC_TO_LDS_B128,
// tracked by ASYNCcnt) with double buffering when the builtins exist; falls
// back to the proven global_load_b128 + ds_store path otherwise.
// ---------------------------------------------------------------------------

typedef __attribute__((ext_vector_type(2))) float v2f;
typedef __attribute__((ext_vector_type(8))) float v8f;
typedef __attribute__((ext_vector_type(4))) float f32x4;

#define BM 64
#define BN 128
#define BK 16
#define LDA_P 20    // floats: 80B row stride (16B aligned for async b128)
#define LDB_P 132   // floats: 528B row stride (16B aligned for async b128)

#if __has_builtin(__builtin_amdgcn_global_load_async_to_lds_b128) && \
    __has_builtin(__builtin_amdgcn_s_wait_asynccnt)
#define USE_ASYNC_LDS 1
#define NBUF 2
#else
#define USE_ASYNC_LDS 0
#define NBUF 1
#endif

#if USE_ASYNC_LDS
// Builtin signature (from hipcc diagnostic): param0 is a pointer to a 16-byte
// int vector: (v4i as1* src, v4i as3* dst, imm offset, imm cpol).
typedef __attribute__((__vector_size__(4 * sizeof(int)))) int i32x4;
typedef __attribute__((address_space(1))) i32x4 gi32x4;
typedef __attribute__((address_space(3))) i32x4 li32x4;

__device__ __forceinline__ void async_copy16(const float* g, float* l) {
    __builtin_amdgcn_global_load_async_to_lds_b128(
        (gi32x4*)(const_cast<float*>(g)),
        (li32x4*)(l),
        /*offset=*/0, /*cpol=*/0);
}
#endif

// EPILOGUE 0: C = acc + bias
// EPILOGUE 1: C = C*w0 + (acc + bias)*w1   (w from resw[row][0..1])
template <int EPILOGUE>
__global__ __launch_bounds__(256, 2)
void wmma_gemm_f32(const float* __restrict__ A, int lda, long long aBatchStride,
                   const float* __restrict__ B, int ldb, long long bBatchStride,
                   const float* __restrict__ bias, int biasStride,
                   float* __restrict__ C, int ldc, long long cBatchStride,
                   int M, int N, int K,
                   const float* __restrict__ resw)
{
    __shared__ float As[NBUF][BM][LDA_P];
    __shared__ float Bs[NBUF][BK][LDB_P];

    const int batch = blockIdx.z;
    A += (long long)batch * aBatchStride;
    B += (long long)batch * bBatchStride;
    C += (long long)batch * cBatchStride;
    if (bias) bias += (long long)batch * biasStride;

    const int mBase = blockIdx.y * BM;
    const int nBase = blockIdx.x * BN;

    const int tid      = threadIdx.x;
    const int lane     = tid & 31;
    const int wave     = tid >> 5;         // 0..7
    const int waveM    = (wave >> 2) * 32; // 0 or 32
    const int waveN    = (wave & 3) * 32;  // 0,32,64,96
    const int laneHalf = lane >> 4;        // 0 or 1
    const int laneMod  = lane & 15;

    v8f acc[2][2] = {};   // 2x2 tiles of 16x16 f32 accumulators

    // Global->LDS staging coordinates (one b128 for A, two for B per thread)
    const int aRow  = tid >> 2;          // 0..63
    const int aCol4 = (tid & 3) * 4;     // 0,4,8,12
    const int bRow0 = tid >> 5;          // 0..7 (+8 for second chunk)
    const int bCol4 = (tid & 31) * 4;    // 0..124

    const float* Aptr = A + (long long)(mBase + aRow) * lda + aCol4;
    const float* Bptr = B + nBase + bCol4;

#if USE_ASYNC_LDS
    // Issue the 3 async DMA copies for K-tile `kt` into buffer `buf`.
    auto issue_tile = [&](int buf, int kt) {
        async_copy16(Aptr + kt, &As[buf][aRow][aCol4]);
        async_copy16(Bptr + (long long)(kt + bRow0) * ldb,     &Bs[buf][bRow0][bCol4]);
        async_copy16(Bptr + (long long)(kt + bRow0 + 8) * ldb, &Bs[buf][bRow0 + 8][bCol4]);
    };
    issue_tile(0, 0);
#endif

    int p = 0;
    for (int kt = 0; kt < K; kt += BK, p ^= (NBUF - 1)) {
#if USE_ASYNC_LDS
        // Software pipeline: kick off next tile's DMA, then wait for the
        // current tile (its 3 ops are the oldest; async loads complete in
        // order, so ASYNCcnt <= 3 means the current tile has landed).
        if (kt + BK < K) {
            issue_tile(p ^ 1, kt + BK);
            __builtin_amdgcn_s_wait_asynccnt(3);
        } else {
            __builtin_amdgcn_s_wait_asynccnt(0);
        }
#else
        // Fallback: stage through VGPRs, with speculative next-tile prefetch.
        if (kt + BK < K) {
            __builtin_prefetch(Aptr + kt + BK, 0, 3);
            __builtin_prefetch(Bptr + (long long)(kt + BK + bRow0) * ldb, 0, 3);
        }
        {
            f32x4 av = *(const f32x4*)(Aptr + kt);
            As[0][aRow][aCol4 + 0] = av.x;
            As[0][aRow][aCol4 + 1] = av.y;
            As[0][aRow][aCol4 + 2] = av.z;
            As[0][aRow][aCol4 + 3] = av.w;
        }
#pragma unroll
        for (int i = 0; i < 2; ++i) {
            int br = bRow0 + i * 8;
            f32x4 bv = *(const f32x4*)(Bptr + (long long)(kt + br) * ldb);
            Bs[0][br][bCol4 + 0] = bv.x;
            Bs[0][br][bCol4 + 1] = bv.y;
            Bs[0][br][bCol4 + 2] = bv.z;
            Bs[0][br][bCol4 + 3] = bv.w;
        }
#endif
        __syncthreads();

        // ---- compute: 4 k-steps of V_WMMA_F32_16X16X4_F32
#pragma unroll
        for (int kk = 0; kk < BK; kk += 4) {
            v2f aF[2], bF[2];
            const int kfrag = kk + laneHalf * 2;
#pragma unroll
            for (int tm = 0; tm < 2; ++tm) {
                // A 16x4 layout: M = lane%16; VGPR0/1 hold K={0,1} (lanes
                // 0-15) or K={2,3} (lanes 16-31)
                int r = waveM + tm * 16 + laneMod;
                aF[tm].x = As[p][r][kfrag];
                aF[tm].y = As[p][r][kfrag + 1];
            }
#pragma unroll
            for (int tn = 0; tn < 2; ++tn) {
                // B 4x16 layout: N = lane%16, K split across lane halves
                int c = waveN + tn * 16 + laneMod;
                bF[tn].x = Bs[p][kfrag][c];
                bF[tn].y = Bs[p][kfrag + 1][c];
            }
#pragma unroll
            for (int tm = 0; tm < 2; ++tm)
#pragma unroll
                for (int tn = 0; tn < 2; ++tn)
                    acc[tm][tn] = __builtin_amdgcn_wmma_f32_16x16x4_f32(
                        false, aF[tm], false, bF[tn],
                        (short)0, acc[tm][tn], false, false);
        }
        __syncthreads();   // protects buffer reuse (t+2 DMA / next sync load)
    }

    // ---- epilogue: C/D layout: VGPR r -> M = r + 8*laneHalf, N = laneMod
#pragma unroll
    for (int tm = 0; tm < 2; ++tm) {
#pragma unroll
        for (int tn = 0; tn < 2; ++tn) {
            const int gn = nBase + waveN + tn * 16 + laneMod;
            const float bv = bias ? bias[gn] : 0.0f;
#pragma unroll
            for (int r = 0; r < 8; ++r) {
                const int gm = mBase + waveM + tm * 16 + laneHalf * 8 + r;
                const long long off = (long long)gm * ldc + gn;
                const float val = acc[tm][tn][r] + bv;
                if (EPILOGUE == 0) {
                    C[off] = val;
                } else {
                    const float w0 = resw[gm * 2 + 0];
                    const float w1 = resw[gm * 2 + 1];
                    C[off] = C[off] * w0 + val * w1;
                }
            }
        }
    }
}

extern "C" void kernel_launch(void* const* d_in, const int* in_sizes, int n_in,
                              void* d_out, int out_size, void* d_ws, size_t ws_size,
                              hipStream_t stream)
{
    const float* inputs          = (const float*)d_in[0]; // (E*C + T, D_IN)
    const float* expert_w        = (const float*)d_in[1]; // (E, D_IN, D_OUT)
    const float* expert_b        = (const float*)d_in[2]; // (E, D_OUT)
    const float* residual_w      = (const float*)d_in[3]; // (D_IN, D_OUT)
    const float* residual_b      = (const float*)d_in[4]; // (D_OUT,)
    const float* combine_w       = (const float*)d_in[5]; // (T, E, C) == (T, E*C)
    const float* residual_weight = (const float*)d_in[6]; // (T, 2)
    float* out = (float*)d_out;                           // (T, D_OUT)

    const int E = 8, Cc = 1024, D_IN = 4096, D_OUT = 1024, T = 4096;

    float* expert_out = (float*)d_ws;  // E*Cc*D_OUT floats = 33.5 MB

    // 1) per-expert GEMM + bias -> expert_out  (batched over grid.z)
    {
        dim3 grid(D_OUT / BN, Cc / BM, E);
        wmma_gemm_f32<0><<<grid, 256, 0, stream>>>(
            inputs, D_IN, (long long)Cc * D_IN,
            expert_w, D_OUT, (long long)D_IN * D_OUT,
            expert_b, D_OUT,
            expert_out, D_OUT, (long long)Cc * D_OUT,
            Cc, D_OUT, D_IN, nullptr);
    }

    // 2) combine GEMM: (T x E*C) @ (E*C x D_OUT) -> out (acts as "combined")
    {
        dim3 grid(D_OUT / BN, T / BM, 1);
        wmma_gemm_f32<0><<<grid, 256, 0, stream>>>(
            combine_w, E * Cc, 0,
            expert_out, D_OUT, 0,
            nullptr, 0,
            out, D_OUT, 0,
            T, D_OUT, E * Cc, nullptr);
    }

    // 3) residual GEMM + blend: out = out*w0 + (resid @ residual_w + rb)*w1
    {
        dim3 grid(D_OUT / BN, T / BM, 1);
        wmma_gemm_f32<1><<<grid, 256, 0, stream>>>(
            inputs + (long long)E * Cc * D_IN, D_IN, 0,
            residual_w, D_OUT, 0,
            residual_b, 0,
            out, D_OUT, 0,
            T, D_OUT, D_IN, residual_weight);
    }
}